// Encoder_49280454754452
// MI455X (gfx1250) — compile-verified
//
#include <hip/hip_runtime.h>
#include <hip/hip_bf16.h>

typedef float v2f __attribute__((ext_vector_type(2)));
typedef float v8f __attribute__((ext_vector_type(8)));

#define FEAT   256          // feature dim
#define KDIM   512          // concat dim (feat + embed)
#define EDIM   256          // output embed dim
#define NSAMP  25           // neighbors per node
#define BT     32           // batch tile per workgroup
#define LDSTR  516          // padded LDS row stride in floats (516 % 64 == 4 -> conflict-free)
#define NAN_FILL 0.01f

__global__ __launch_bounds__(256)
void encoder_sage_fused(const float* __restrict__ features,
                        const float* __restrict__ weight,
                        const long long* __restrict__ nodes,
                        const long long* __restrict__ neigh,
                        float* __restrict__ out,
                        int batch)
{
    __shared__ float comb[BT * LDSTR];     // combined tile [32][512] (+4 pad/row) ~64.5 KB
    __shared__ int   sIdx[BT][NSAMP];      // neighbor indices for this tile
    __shared__ int   sNode[BT];            // self indices

    const int tid   = threadIdx.x;
    const int bbase = blockIdx.x * BT;

    // ---------------- index staging ----------------
    for (int i = tid; i < BT * NSAMP; i += 256) {
        const int r = i / NSAMP, s = i - r * NSAMP;
        const int b = bbase + r;
        sIdx[r][s] = (b < batch) ? (int)neigh[(long long)b * NSAMP + s] : -1;
    }
    if (tid < BT) {
        const int b = bbase + tid;
        sNode[tid] = (b < batch) ? (int)nodes[b] : -1;
    }
    __syncthreads();

    // ---------------- stage 1: build combined tile in LDS ----------------
    // 8 threads per row; each covers 32 columns (8 x float4) of self and of neigh-mean.
    {
        const int r  = tid >> 3;   // 0..31  (node row in tile)
        const int c8 = tid & 7;    // 0..7   (column chunk)
        const int node = sNode[r];
        const float4* __restrict__ feat4 = (const float4*)features;
        float* dstSelf  = &comb[r * LDSTR +        c8 * 32];
        float* dstNeigh = &comb[r * LDSTR + FEAT + c8 * 32];

        if (node >= 0) {
            const float4* srow = feat4 + (size_t)node * (FEAT / 4) + c8 * 8;
            #pragma unroll
            for (int j = 0; j < 8; ++j)
                *(float4*)(dstSelf + j * 4) = srow[j];

            const float invn = 1.0f / (float)NSAMP;
            #pragma unroll
            for (int j = 0; j < 8; ++j) {
                float4 acc = make_float4(0.f, 0.f, 0.f, 0.f);
                for (int s = 0; s < NSAMP; ++s) {
                    const int ni = sIdx[r][s];
                    const float4 v = feat4[(size_t)ni * (FEAT / 4) + c8 * 8 + j];
                    acc.x += v.x; acc.y += v.y; acc.z += v.z; acc.w += v.w;
                }
                acc.x *= invn; acc.y *= invn; acc.z *= invn; acc.w *= invn;
                acc.x = (acc.x != acc.x) ? NAN_FILL : acc.x;
                acc.y = (acc.y != acc.y) ? NAN_FILL : acc.y;
                acc.z = (acc.z != acc.z) ? NAN_FILL : acc.z;
                acc.w = (acc.w != acc.w) ? NAN_FILL : acc.w;
                *(float4*)(dstNeigh + j * 4) = acc;
            }
        } else {
            const float4 z = make_float4(0.f, 0.f, 0.f, 0.f);
            #pragma unroll
            for (int j = 0; j < 8; ++j) {
                *(float4*)(dstSelf  + j * 4) = z;
                *(float4*)(dstNeigh + j * 4) = z;
            }
        }
    }
    __syncthreads();

    // ---------------- stage 2: fp32 WMMA GEMM  out = relu(W @ combined^T) ----------------
    // Wave w computes E rows [32w, 32w+32) x B cols [bbase, bbase+32) as 4 wmma 16x16 tiles.
    const int lane = tid & 31;
    const int wave = tid >> 5;             // 0..7
    const int m    = lane & 15;            // A row / B col within 16
    const int kh   = (lane >> 4) << 1;     // lanes 0-15 -> K {k,k+1}; lanes 16-31 -> {k+2,k+3}
    const int e0   = wave * 32;

    const float* wrow0 = weight + (size_t)(e0 + m)      * KDIM + kh;   // A tile 0
    const float* wrow1 = weight + (size_t)(e0 + 16 + m) * KDIM + kh;   // A tile 1
    const float* brow0 = &comb[(0  + m) * LDSTR + kh];                 // B tile 0 (LDS)
    const float* brow1 = &comb[(16 + m) * LDSTR + kh];                 // B tile 1 (LDS)

    v8f acc00 = {}; v8f acc01 = {}; v8f acc10 = {}; v8f acc11 = {};

    #pragma unroll 4
    for (int k = 0; k < KDIM; k += 4) {
        const v2f a0 = *(const v2f*)(wrow0 + k);
        const v2f a1 = *(const v2f*)(wrow1 + k);
        const v2f b0 = *(const v2f*)(brow0 + k);
        const v2f b1 = *(const v2f*)(brow1 + k);
        acc00 = __builtin_amdgcn_wmma_f32_16x16x4_f32(false, a0, false, b0, (short)0, acc00, false, false);
        acc01 = __builtin_amdgcn_wmma_f32_16x16x4_f32(false, a0, false, b1, (short)0, acc01, false, false);
        acc10 = __builtin_amdgcn_wmma_f32_16x16x4_f32(false, a1, false, b0, (short)0, acc10, false, false);
        acc11 = __builtin_amdgcn_wmma_f32_16x16x4_f32(false, a1, false, b1, (short)0, acc11, false, false);
    }

    // ---------------- ReLU + store ----------------
    // C/D layout: VGPR r, lane l -> M = r + (l>=16 ? 8 : 0), N = l & 15
    const int nlo = lane & 15;
    const int mhi = (lane >> 4) * 8;

    {
        const int bc0 = bbase + nlo;        // tiles *0
        const int bc1 = bbase + 16 + nlo;   // tiles *1
        if (bc0 < batch) {
            #pragma unroll
            for (int r = 0; r < 8; ++r) {
                float v0 = acc00[r]; v0 = v0 > 0.f ? v0 : 0.f;
                float v1 = acc10[r]; v1 = v1 > 0.f ? v1 : 0.f;
                out[(size_t)(e0      + mhi + r) * batch + bc0] = v0;
                out[(size_t)(e0 + 16 + mhi + r) * batch + bc0] = v1;
            }
        }
        if (bc1 < batch) {
            #pragma unroll
            for (int r = 0; r < 8; ++r) {
                float v0 = acc01[r]; v0 = v0 > 0.f ? v0 : 0.f;
                float v1 = acc11[r]; v1 = v1 > 0.f ? v1 : 0.f;
                out[(size_t)(e0      + mhi + r) * batch + bc1] = v0;
                out[(size_t)(e0 + 16 + mhi + r) * batch + bc1] = v1;
            }
        }
    }
}

extern "C" void kernel_launch(void* const* d_in, const int* in_sizes, int n_in,
                              void* d_out, int out_size, void* d_ws, size_t ws_size,
                              hipStream_t stream) {
    const float*     features = (const float*)d_in[0];      // [100000, 256] fp32
    const float*     weight   = (const float*)d_in[1];      // [256, 512]    fp32
    const long long* nodes    = (const long long*)d_in[2];  // [B]   int64
    const long long* neigh    = (const long long*)d_in[3];  // [B,25] int64
    float*           out      = (float*)d_out;              // [256, B] fp32

    const int batch = in_sizes[2];                          // 50000
    const int grid  = (batch + BT - 1) / BT;

    encoder_sage_fused<<<grid, 256, 0, stream>>>(features, weight, nodes, neigh, out, batch);
}